// DecoderRNN_36919538877107
// MI455X (gfx1250) — compile-verified
//
#include <hip/hip_runtime.h>
#include <hip/hip_bf16.h>
#include <math.h>

// ---------------- problem constants ----------------
#define SEQ   2048
#define BATCH 64
#define HID   512
#define EMB   256
#define VOC   50257
#define G3H   1536           // 3*HID
#define RSQRT_H 0.044194173824159216f  // 1/sqrt(512)

typedef __attribute__((ext_vector_type(2))) float v2f;
typedef __attribute__((ext_vector_type(8))) float v8f;

// =====================================================================
// Kernel 1: attention energy + softmax (one block per batch element)
// =====================================================================
__global__ __launch_bounds__(256)
void attn_energy_softmax(const float* __restrict__ dec_hidden,  // [2,B,H]
                         const float* __restrict__ enc,         // [S,B,H]
                         float* __restrict__ alpha,             // [B,S]
                         float* __restrict__ context)           // [B,H] (zeroed here)
{
    __shared__ float htop[HID];
    __shared__ float ener[SEQ];
    __shared__ float red[8];

    const int b    = blockIdx.x;
    const int tid  = threadIdx.x;
    const int wave = tid >> 5;
    const int lane = tid & 31;

    // zero the context accumulator for kernel 2
    context[b * HID + tid]       = 0.0f;
    context[b * HID + 256 + tid] = 0.0f;

    // stage h_top = dec_hidden[1][b]
    htop[tid]       = dec_hidden[(size_t)BATCH * HID + (size_t)b * HID + tid];
    htop[tid + 256] = dec_hidden[(size_t)BATCH * HID + (size_t)b * HID + tid + 256];
    __syncthreads();

    // energy: 8 waves, each wave handles 256 seq positions
    for (int si = 0; si < SEQ / 8; ++si) {
        const int s = si * 8 + wave;
        const float* e = enc + ((size_t)s * BATCH + b) * HID;
        float sum = 0.0f;
        #pragma unroll
        for (int h = lane; h < HID; h += 32) sum += htop[h] * e[h];
        #pragma unroll
        for (int off = 16; off > 0; off >>= 1) sum += __shfl_xor(sum, off, 32);
        if (lane == 0) ener[s] = sum * RSQRT_H;
    }
    __syncthreads();

    // ---- softmax over SEQ in LDS ----
    float m = -INFINITY;
    for (int s = tid; s < SEQ; s += 256) m = fmaxf(m, ener[s]);
    #pragma unroll
    for (int off = 16; off > 0; off >>= 1) m = fmaxf(m, __shfl_xor(m, off, 32));
    if (lane == 0) red[wave] = m;
    __syncthreads();
    m = red[0];
    #pragma unroll
    for (int i = 1; i < 8; ++i) m = fmaxf(m, red[i]);
    __syncthreads();

    float lsum = 0.0f;
    for (int s = tid; s < SEQ; s += 256) {
        float v = __expf(ener[s] - m);
        ener[s] = v;
        lsum += v;
    }
    #pragma unroll
    for (int off = 16; off > 0; off >>= 1) lsum += __shfl_xor(lsum, off, 32);
    if (lane == 0) red[wave] = lsum;
    __syncthreads();
    float tot = 0.0f;
    #pragma unroll
    for (int i = 0; i < 8; ++i) tot += red[i];
    const float inv = 1.0f / tot;

    for (int s = tid; s < SEQ; s += 256)
        alpha[(size_t)b * SEQ + s] = ener[s] * inv;
}

// =====================================================================
// Kernel 2: context[b][h] += sum_{s in chunk} alpha[s] * enc[s][b][h]
// =====================================================================
__global__ __launch_bounds__(256)
void attn_context(const float* __restrict__ alpha,   // [B,S]
                  const float* __restrict__ enc,     // [S,B,H]
                  float* __restrict__ context)       // [B,H]
{
    const int b   = blockIdx.x;
    const int sc  = blockIdx.y;        // seq chunk
    const int tid = threadIdx.x;

    __shared__ float al[256];
    al[tid] = alpha[(size_t)b * SEQ + sc * 256 + tid];
    __syncthreads();

    float acc0 = 0.0f, acc1 = 0.0f;
    const float* base = enc + ((size_t)(sc * 256) * BATCH + b) * HID;
    for (int si = 0; si < 256; ++si) {
        const float* e = base + (size_t)si * BATCH * HID;
        const float a = al[si];
        acc0 += a * e[tid];
        acc1 += a * e[tid + 256];
    }
    atomicAdd(&context[b * HID + tid],       acc0);
    atomicAdd(&context[b * HID + 256 + tid], acc1);
}

// =====================================================================
// Kernel 3: x[b] = concat(emb[inputs[b]], context[b])   -> [B, EMB+HID]
// =====================================================================
__global__ __launch_bounds__(256)
void embed_concat(const int* __restrict__ inputs,
                  const float* __restrict__ emb,       // [V,E]
                  const float* __restrict__ context,   // [B,H]
                  float* __restrict__ x)               // [B, E+H]
{
    const int b   = blockIdx.x;
    const int tid = threadIdx.x;
    const int tok = inputs[b];
    x[(size_t)b * (EMB + HID) + tid]             = emb[(size_t)tok * EMB + tid];
    x[(size_t)b * (EMB + HID) + EMB + tid]       = context[b * HID + tid];
    x[(size_t)b * (EMB + HID) + EMB + 256 + tid] = context[b * HID + 256 + tid];
}

// =====================================================================
// Kernel 4: generic fp32 WMMA GEMM, fixed M=64:
//   C[64, N] = A[64, K] @ W[N, K]^T + bias[N]
//   - 64x64 output tile per block, 8 waves x 2 (16x16) WMMA tiles
//   - W tile double-buffered in LDS via async global->LDS b128 copies
//     (ASYNCcnt pipelined: issue chunk c+1, wait <=4 for chunk c, barrier)
//   - row stride 68 floats => conflict-free WMMA B reads, 16B alignment
//   - K must be a multiple of 64; N tail handled by row clamping (the
//     clamped B columns only feed output columns >= N, never stored),
//     keeping EXEC all-1s as WMMA requires.
// =====================================================================
#define GEMM_KC 64

__global__ __launch_bounds__(256)
void gemm64_wmma(const float* __restrict__ A,
                 const float* __restrict__ W,
                 const float* __restrict__ bias,
                 float* __restrict__ C,
                 int K, int N, int ldc)
{
    __shared__ __align__(16) float wtile[2][64][GEMM_KC + 4];  // 2 x 17KB

    const int n0   = blockIdx.x * 64;
    const int tid  = threadIdx.x;
    const int wave = tid >> 5;
    const int lane = tid & 31;

    const int mt    = (wave & 3) * 16;           // row tile (M offset)
    const int ct0   = ((wave >> 2) * 2) * 16;    // first col tile
    const int ct1   = ct0 + 16;
    const int lm    = lane & 15;                 // m (and n) within 16
    const int khalf = (lane >> 4) * 2;           // 0 or 2

    v8f acc0 = {0.f,0.f,0.f,0.f,0.f,0.f,0.f,0.f};
    v8f acc1 = {0.f,0.f,0.f,0.f,0.f,0.f,0.f,0.f};

    // issue one chunk of async global->LDS copies (4 x b128 per thread)
    auto issue_chunk = [&](int k0c, int bufsel) {
        #pragma unroll
        for (int i = 0; i < 4; ++i) {
            const int idx   = tid + i * 256;     // float4 index 0..1023
            const int row   = idx >> 4;
            const int colf4 = idx & 15;
            int gRow = n0 + row;
            if (gRow >= N) gRow = N - 1;         // clamp: keeps EXEC all-1s
            const float* src = W + (size_t)gRow * K + k0c + colf4 * 4;
            const unsigned ldsb =
                (unsigned)(unsigned long long)&wtile[bufsel][row][colf4 * 4];
            asm volatile("global_load_async_to_lds_b128 %0, %1, off"
                         :: "v"(ldsb), "v"(src) : "memory");
        }
    };

    const int nc = K / GEMM_KC;
    issue_chunk(0, 0);

    for (int c = 0; c < nc; ++c) {
        if (c + 1 < nc) {
            issue_chunk((c + 1) * GEMM_KC, (c + 1) & 1);
            // 4 newest async ops are the prefetch; <=4 means chunk c landed
            asm volatile("s_wait_asynccnt 0x4" ::: "memory");
        } else {
            asm volatile("s_wait_asynccnt 0x0" ::: "memory");
        }
        __syncthreads();   // chunk c visible to all waves

        const float (*wt)[GEMM_KC + 4] = wtile[c & 1];
        const float* arow = A + (size_t)(mt + lm) * K + c * GEMM_KC;
        #pragma unroll 4
        for (int k = 0; k < GEMM_KC; k += 4) {
            v2f a  = *(const v2f*)(arow + k + khalf);
            v2f b0 = *(const v2f*)&wt[ct0 + lm][k + khalf];
            v2f b1 = *(const v2f*)&wt[ct1 + lm][k + khalf];
            acc0 = __builtin_amdgcn_wmma_f32_16x16x4_f32(
                       false, a, false, b0, (short)0, acc0, false, false);
            acc1 = __builtin_amdgcn_wmma_f32_16x16x4_f32(
                       false, a, false, b1, (short)0, acc1, false, false);
        }
        __syncthreads();   // all waves done reading buf (c&1) before reuse
    }

    // ---- store D (+bias) ----
    const int mrow = mt + ((lane >> 4) * 8);
    #pragma unroll
    for (int r = 0; r < 8; ++r) {
        const int m = mrow + r;
        int n = n0 + ct0 + lm;
        if (n < N) C[(size_t)m * ldc + n] = acc0[r] + bias[n];
        n = n0 + ct1 + lm;
        if (n < N) C[(size_t)m * ldc + n] = acc1[r] + bias[n];
    }
}

// =====================================================================
// Kernel 5: fused GRU gates (PyTorch order r,z,n): h = (1-z)*n + z*h_prev
// =====================================================================
__global__ __launch_bounds__(256)
void gru_gate(const float* __restrict__ gi,     // [B,3H]
              const float* __restrict__ gh,     // [B,3H]
              const float* __restrict__ hprev,  // [B,H]
              float* __restrict__ hout)         // [B,H]
{
    const int idx = blockIdx.x * blockDim.x + threadIdx.x;   // < B*H
    const int b = idx >> 9;
    const int j = idx & (HID - 1);
    const float* gib = gi + (size_t)b * G3H;
    const float* ghb = gh + (size_t)b * G3H;
    const float r = 1.0f / (1.0f + __expf(-(gib[j]           + ghb[j])));
    const float z = 1.0f / (1.0f + __expf(-(gib[HID + j]     + ghb[HID + j])));
    const float n = tanhf(gib[2 * HID + j] + r * ghb[2 * HID + j]);
    hout[idx] = (1.0f - z) * n + z * hprev[idx];
}

// =====================================================================
// Host side
// =====================================================================
extern "C" void kernel_launch(void* const* d_in, const int* in_sizes, int n_in,
                              void* d_out, int out_size, void* d_ws, size_t ws_size,
                              hipStream_t stream) {
    const int*   inputs     = (const int*)  d_in[0];
    const float* dec_hidden = (const float*)d_in[1];   // [2,B,H]
    const float* enc        = (const float*)d_in[2];   // [S,B,H]
    const float* emb        = (const float*)d_in[3];   // [V,E]
    const float* W_ih0      = (const float*)d_in[4];   // [3H, H+E]
    const float* W_hh0      = (const float*)d_in[5];   // [3H, H]
    const float* b_ih0      = (const float*)d_in[6];
    const float* b_hh0      = (const float*)d_in[7];
    const float* W_ih1      = (const float*)d_in[8];   // [3H, H]
    const float* W_hh1      = (const float*)d_in[9];   // [3H, H]
    const float* b_ih1      = (const float*)d_in[10];
    const float* b_hh1      = (const float*)d_in[11];
    const float* out_W      = (const float*)d_in[12];  // [V, H]
    const float* out_b      = (const float*)d_in[13];

    float* out    = (float*)d_out;
    float* logits = out;                                  // [B, V]
    float* h0_out = out + (size_t)BATCH * VOC;            // [B, H]
    float* h1_out = h0_out + (size_t)BATCH * HID;         // [B, H]

    // workspace layout (floats)
    float* ws      = (float*)d_ws;
    float* alpha   = ws;                                  // B*SEQ   = 131072
    float* context = alpha + (size_t)BATCH * SEQ;         // B*H     =  32768
    float* x       = context + (size_t)BATCH * HID;       // B*(E+H) =  49152
    float* gi      = x + (size_t)BATCH * (EMB + HID);     // B*3H    =  98304
    float* gh      = gi + (size_t)BATCH * G3H;            // B*3H    =  98304

    const float* hprev0 = dec_hidden;                       // layer 0 hidden
    const float* hprev1 = dec_hidden + (size_t)BATCH * HID; // layer 1 hidden (= h_top)

    // 1) attention energy + softmax (also zeroes context accumulator)
    attn_energy_softmax<<<BATCH, 256, 0, stream>>>(dec_hidden, enc, alpha, context);

    // 2) context accumulation
    attn_context<<<dim3(BATCH, 8), 256, 0, stream>>>(alpha, enc, context);

    // 3) embed + concat
    embed_concat<<<BATCH, 256, 0, stream>>>(inputs, emb, context, x);

    // 4) GRU layer 0 gates
    gemm64_wmma<<<G3H / 64, 256, 0, stream>>>(x,      W_ih0, b_ih0, gi, EMB + HID, G3H, G3H);
    gemm64_wmma<<<G3H / 64, 256, 0, stream>>>(hprev0, W_hh0, b_hh0, gh, HID,       G3H, G3H);
    gru_gate<<<(BATCH * HID) / 256, 256, 0, stream>>>(gi, gh, hprev0, h0_out);

    // 5) GRU layer 1
    gemm64_wmma<<<G3H / 64, 256, 0, stream>>>(h0_out, W_ih1, b_ih1, gi, HID, G3H, G3H);
    gemm64_wmma<<<G3H / 64, 256, 0, stream>>>(hprev1, W_hh1, b_hh1, gh, HID, G3H, G3H);
    gru_gate<<<(BATCH * HID) / 256, 256, 0, stream>>>(gi, gh, hprev1, h1_out);

    // 6) vocab projection: logits = h1 @ out_W^T + out_b   (N = 50257, tail-guarded)
    gemm64_wmma<<<(VOC + 63) / 64, 256, 0, stream>>>(h1_out, out_W, out_b,
                                                     logits, HID, VOC, VOC);
}